// Quanv1D_23081154248819
// MI455X (gfx1250) — compile-verified
//
#include <hip/hip_runtime.h>
#include <math.h>

// ---------------------------------------------------------------------------
// Quanv1D on MI455X (gfx1250):
//   final_state = V (fixed 256x256 complex unitary) x product_state(patch)
//   -> complex GEMM (M=65480, N=K=256) on V_WMMA_F32_16X16X4_F32,
//      probs/expectation epilogue fused in-register.
// Operands packed so each K-step needs exactly one ds_load_b128 (A re+im pair)
// and one global_load_b128 (B re+im pair) feeding 4 WMMAs.
// ---------------------------------------------------------------------------

typedef __attribute__((ext_vector_type(2))) float v2f;
typedef __attribute__((ext_vector_type(4))) float v4f;
typedef __attribute__((ext_vector_type(8))) float v8f;

#define NQ     8
#define DIM    256            // 2^NQ amplitudes
#define KWIN   8
#define OUTC   8
#define LEN    8192
#define LOUT   (LEN - KWIN + 1)     // 8185
#define BATCH  8
#define NPATCH (BATCH * LOUT)       // 65480
#define TILE_M 16
#define SROWF  520                  // padded LDS row stride in floats (16B-aligned rows)
#define PI_F   3.14159265358979f

// ---------------------------------------------------------------------------
// Kernel 1: build packed Vq from V = CNOT2 o RotL2 o CNOT1.
//   Vq[((k>>1)*256 + n)*4 + {0,1,2,3}] = {Vre[k][n], Vre[k+1][n],
//                                         Vim[k][n], Vim[k+1][n]},  k = column j.
// One block per input basis column j; 256 threads simulate the statevector.
// CNOT rings are pure basis-index permutations (folded into init / writeout).
// ---------------------------------------------------------------------------
__global__ void __launch_bounds__(256)
build_unitary(const float* __restrict__ weights, float* __restrict__ vq) {
  __shared__ float sre[DIM];
  __shared__ float sim[DIM];
  const int j = blockIdx.x;   // input basis index (row k of Vt)
  const int i = threadIdx.x;  // amplitude index this thread owns

  // perm1: CNOT ring r=1, e_j -> e_{perm1(j)}. wire w <-> bit (7-w).
  int j1 = j;
#pragma unroll
  for (int w = 0; w < NQ; ++w) {
    int t = (w + 1) & 7;
    if ((j1 >> (7 - w)) & 1) j1 ^= 1 << (7 - t);
  }
  sre[i] = (i == j1) ? 1.0f : 0.0f;
  sim[i] = 0.0f;
  __syncthreads();

  // Rot layer, weights layer l=1 (second layer). All threads compute the same
  // 2x2 matrix (cheap), each thread updates its own amplitude.
#pragma unroll 1
  for (int w = 0; w < NQ; ++w) {
    const float phi = weights[24 + w * 3 + 0];
    const float th  = weights[24 + w * 3 + 1];
    const float om  = weights[24 + w * 3 + 2];
    const float ct = cosf(0.5f * th), st = sinf(0.5f * th);
    const float a  = 0.5f * (phi + om), b = 0.5f * (phi - om);
    const float ca = cosf(a), sa = sinf(a), cb = cosf(b), sb = sinf(b);
    // m00 = e^{-ia} ct ; m01 = -e^{+ib} st ; m10 = e^{-ib} st ; m11 = e^{+ia} ct
    const float m00r =  ca * ct, m00i = -sa * ct;
    const float m01r = -cb * st, m01i = -sb * st;
    const float m10r =  cb * st, m10i = -sb * st;
    const float m11r =  ca * ct, m11i =  sa * ct;

    const int mask    = 1 << (7 - w);
    const int partner = i ^ mask;
    const float ar = sre[i],       ai = sim[i];
    const float br = sre[partner], bi = sim[partner];
    __syncthreads();
    float nr, ni;
    if ((i & mask) == 0) {      // |0> slot: new = m00*self + m01*partner
      nr = m00r * ar - m00i * ai + m01r * br - m01i * bi;
      ni = m00r * ai + m00i * ar + m01r * bi + m01i * br;
    } else {                    // |1> slot: new = m10*partner + m11*self
      nr = m10r * br - m10i * bi + m11r * ar - m11i * ai;
      ni = m10r * bi + m10i * br + m11r * ai + m11i * ar;
    }
    sre[i] = nr;
    sim[i] = ni;
    __syncthreads();
  }

  // perm2: CNOT ring r=2 applied to a general state: new[perm2(i)] = old[i].
  int i2 = i;
#pragma unroll
  for (int w = 0; w < NQ; ++w) {
    int t = (w + 2) & 7;
    if ((i2 >> (7 - w)) & 1) i2 ^= 1 << (7 - t);
  }
  // Packed interleaved writeout: pair index = j>>1, slot = j&1 (re), +2 (im).
  const int base = ((j >> 1) * DIM + i2) * 4 + (j & 1);
  vq[base]     = sre[i];
  vq[base + 2] = sim[i];
}

// ---------------------------------------------------------------------------
// Kernel 2: one wave32 per 16-patch tile.
// Phase 1: build 16 product states (data gate folded with Rot layer l=0) into
//          LDS, packed as {re_k, re_k+1, im_k, im_k+1} per even-k pair.
// Phase 2: F = S * Vt via f32 WMMA (complex: 4 chains over 2 accumulators),
//          epilogue: probs = |F|^2, e_o = sum_n sign_o(n) probs[n], + bias.
// ---------------------------------------------------------------------------
__global__ void __launch_bounds__(32)
quanv_main(const float* __restrict__ x, const float* __restrict__ weights,
           const float* __restrict__ bias, const float* __restrict__ vq,
           float* __restrict__ out) {
  __shared__ float SC[TILE_M * SROWF];
  const int lane  = threadIdx.x;
  const int pbase = blockIdx.x * TILE_M;

  // ---------------- Phase 1: product states into LDS ----------------
  {
    const int pi   = lane >> 1;   // patch within tile (2 threads/patch)
    const int half = lane & 1;    // which 128-amplitude half this thread fills
    const int p    = pbase + pi;
    float v0r[NQ], v0i[NQ], v1r[NQ], v1i[NQ];
    if (p < NPATCH) {
      const int b = p / LOUT, l = p % LOUT;
#pragma unroll
      for (int w = 0; w < NQ; ++w) {
        const float xv  = x[b * LEN + l + w];
        const float ang = PI_F * tanhf(xv);
        const float c = cosf(0.5f * ang), s = sinf(0.5f * ang);
        // data gate on |0>: u = (c, -i s)  ->  u0=(c,0), u1=(0,-s)
        const float phi = weights[w * 3 + 0];
        const float th  = weights[w * 3 + 1];
        const float om  = weights[w * 3 + 2];
        const float ct = cosf(0.5f * th), st = sinf(0.5f * th);
        const float a  = 0.5f * (phi + om), bb = 0.5f * (phi - om);
        const float ca = cosf(a), sa = sinf(a), cb = cosf(bb), sb = sinf(bb);
        const float m00r =  ca * ct, m00i = -sa * ct;
        const float m01r = -cb * st, m01i = -sb * st;
        const float m10r =  cb * st, m10i = -sb * st;
        const float m11r =  ca * ct, m11i =  sa * ct;
        // v = RotL0_w * u  (complex 2-vec per qubit)
        v0r[w] = m00r * c + m01i * s;   // m01 * (-i s) = ( m01i*s, -m01r*s )
        v0i[w] = m00i * c - m01r * s;
        v1r[w] = m10r * c + m11i * s;
        v1i[w] = m10i * c - m11r * s;
      }
    } else {
#pragma unroll
      for (int w = 0; w < NQ; ++w) { v0r[w] = v0i[w] = v1r[w] = v1i[w] = 0.0f; }
    }
    const int d0 = half * 128;
    for (int d = d0; d < d0 + 128; ++d) {
      float re = 1.0f, im = 0.0f;
#pragma unroll
      for (int w = 0; w < NQ; ++w) {
        const int   bit = (d >> (7 - w)) & 1;
        const float vr  = bit ? v1r[w] : v0r[w];
        const float vi  = bit ? v1i[w] : v0i[w];
        const float nre = re * vr - im * vi;
        im = re * vi + im * vr;
        re = nre;
      }
      // Packed pair layout: group = d>>1, slot = d&1 (re), +2 (im).
      const int sb = pi * SROWF + (d >> 1) * 4 + (d & 1);
      SC[sb]     = re;
      SC[sb + 2] = im;
    }
  }
  __syncthreads();

  // ---------------- Phase 2: complex GEMM on WMMA f32 ----------------
  // A (16x4 f32): lanes 0-15 row M=lane, K-pair {0,1}; lanes 16-31 row M=lane-16,
  // K-pair {2,3}. B (4x16 f32): lanes 0-15 col N=lane, K rows {0,1}; lanes 16-31
  // col N=lane-16, K rows {2,3}.  K pairs are even-aligned -> one packed float4
  // per lane per matrix pair: {re_k, re_k+1, im_k, im_k+1}.
  const int mrow = lane & 15;
  const int khalf = lane >> 4;        // 0: K{0,1}, 1: K{2,3} within each 4-step

  v8f accs[OUTC];
#pragma unroll
  for (int o = 0; o < OUTC; ++o)
    accs[o] = (v8f){0.f, 0.f, 0.f, 0.f, 0.f, 0.f, 0.f, 0.f};

#pragma unroll 1
  for (int nt = 0; nt < 16; ++nt) {
    v8f cre = (v8f){0.f, 0.f, 0.f, 0.f, 0.f, 0.f, 0.f, 0.f};
    v8f cim = (v8f){0.f, 0.f, 0.f, 0.f, 0.f, 0.f, 0.f, 0.f};
    const int n = nt * 16 + mrow;      // global state index handled by this lane

#pragma unroll 4
    for (int kb = 0; kb < 64; ++kb) {
      const int kp = kb * 2 + khalf;   // even-k pair index (k = 2*kp)
      const v4f ac = *(const v4f*)&SC[mrow * SROWF + kp * 4];        // ds_load_b128
      const v4f bc = *(const v4f*)&vq[(kp * DIM + n) * 4];           // global_load_b128
      v2f are, aim, aimn, bre, bim;
      are.x = ac.x;  are.y = ac.y;
      aim.x = ac.z;  aim.y = ac.w;
      aimn.x = -ac.z; aimn.y = -ac.w;
      bre.x = bc.x;  bre.y = bc.y;
      bim.x = bc.z;  bim.y = bc.w;
      // Fre = Sre*Bre - Sim*Bim ; Fim = Sre*Bim + Sim*Bre
      cre = __builtin_amdgcn_wmma_f32_16x16x4_f32(false, are,  false, bre, (short)0, cre, false, false);
      cre = __builtin_amdgcn_wmma_f32_16x16x4_f32(false, aimn, false, bim, (short)0, cre, false, false);
      cim = __builtin_amdgcn_wmma_f32_16x16x4_f32(false, are,  false, bim, (short)0, cim, false, false);
      cim = __builtin_amdgcn_wmma_f32_16x16x4_f32(false, aim,  false, bre, (short)0, cim, false, false);
    }

    // Epilogue: probs and +/-1 sign contraction (sign is lane-constant per o).
    const v8f probs = cre * cre + cim * cim;
#pragma unroll
    for (int o = 0; o < OUTC; ++o) {
      const float sgn = ((n >> (7 - o)) & 1) ? -1.0f : 1.0f;
      accs[o] += probs * sgn;
    }
  }

  // Reduce across the 16 lanes of each half (lanes share the same 8 patches).
#pragma unroll
  for (int o = 0; o < OUTC; ++o) {
    v8f a = accs[o];
#pragma unroll
    for (int m = 1; m <= 8; m <<= 1) {
      v8f t;
#pragma unroll
      for (int e = 0; e < 8; ++e) t[e] = __shfl_xor(a[e], m, 32);
      a += t;
    }
    accs[o] = a;
  }

  if (mrow == 0) {
    const int prow = (lane >> 4) * 8;   // lane 0 -> patches 0-7, lane 16 -> 8-15
#pragma unroll
    for (int o = 0; o < OUTC; ++o) {
      const float bo = bias[o];
#pragma unroll
      for (int r = 0; r < 8; ++r) {
        const int p = pbase + prow + r;
        if (p < NPATCH) {
          const int b = p / LOUT, l = p % LOUT;
          out[b * (OUTC * LOUT) + o * LOUT + l] = accs[o][r] + bo;
        }
      }
    }
  }
}

// ---------------------------------------------------------------------------
extern "C" void kernel_launch(void* const* d_in, const int* in_sizes, int n_in,
                              void* d_out, int out_size, void* d_ws, size_t ws_size,
                              hipStream_t stream) {
  const float* x       = (const float*)d_in[0];  // (8,1,8192) f32
  const float* weights = (const float*)d_in[1];  // (2,8,3) f32
  const float* bias    = (const float*)d_in[2];  // (8,) f32

  float* vq = (float*)d_ws;   // packed Vt: 128 pairs x 256 n x 4 f32 = 512 KB

  build_unitary<<<DIM, DIM, 0, stream>>>(weights, vq);

  const int ntiles = (NPATCH + TILE_M - 1) / TILE_M;   // 4093
  quanv_main<<<ntiles, 32, 0, stream>>>(x, weights, bias, vq, (float*)d_out);
}